// DiagonalPixelLSTM_45114336477858
// MI455X (gfx1250) — compile-verified
//
#include <hip/hip_runtime.h>
#include <math.h>

// DiagonalPixelLSTM fused kernel for MI455X (gfx1250, wave32, WMMA).
//
// One workgroup per batch image. All recurrent state lives in LDS (~150 KB of
// the 320 KB/WGP). Per diagonal step one fused matmul
//     g = [W_is | W_ss[:,:,1] | W_ss[:,:,0]] (320x192) @ [x_col; h; h<<1] + bias
// via V_WMMA_F32_16X16X4_F32 (exact fp32). 20 waves: wave w owns M-tile w and
// keeps its 16x192 A-slice in 96 VGPRs (loop-invariant across all 127 steps);
// 4 independent accumulator chains (N-tiles) interleave per K-chunk.
//
// LDS layout choices:
//  * x/h transposed, K-contiguous, as v2f arrays -> each WMMA B operand is a
//    single aligned ds_load_b64 straight into an even VGPR pair (no shuffles).
//    Float row stride 68 (== 4 mod 64): banks touched = 4*l16 + 2*hi + e, a
//    perfect permutation of all 64 banks -> conflict-free.
//  * h and c carry a zeroed row 0 (column i stored at row i+1), so the
//    H-shifted reads need no per-lane conditional in the hot WMMA chain.
//  * Next column's global feature gather is issued into registers right after
//    the first barrier, overlapping its latency with matmul + gate math.

typedef float v2f __attribute__((ext_vector_type(2)));
typedef float v8f __attribute__((ext_vector_type(8)));

#define CIN   64
#define HDIM  64
#define WDIM  64
#define HID   64
#define GATES 320            // 5*HID
#define NKC   48             // 192 / 4 (K per fp32 WMMA)
#define WAVES 20             // GATES / 16
#define NT    4              // HDIM / 16
#define NTHREADS 640
#define NLOAD 7              // ceil(64*64 / 640)
#define XS2   34             // v2f row stride for x/h (= 68 floats, 68 % 64 == 4)
#define XSF   68             // float view of the same stride
#define GS    65             // float row stride for g/c

__device__ __forceinline__ float sigf(float x) { return 1.0f / (1.0f + __expf(-x)); }

__device__ __forceinline__ void gather_col(const float* __restrict__ featB, int col,
                                           int tid, float* nx) {
#pragma unroll
    for (int t = 0; t < NLOAD; ++t) {
        const int idx = tid + t * NTHREADS;
        float v = 0.0f;
        if (idx < CIN * HDIM) {
            const int c = idx & 63, i = idx >> 6;
            const int w = col - i;
            if (w >= 0 && w < WDIM) v = featB[(c * HDIM + i) * WDIM + w];
        }
        nx[t] = v;
    }
}

__launch_bounds__(NTHREADS, 1)
__global__ void diag_pixel_lstm(const float* __restrict__ feat,   // [B,C,H,W]
                                const float* __restrict__ W_is,   // [320,64]
                                const float* __restrict__ b_is,   // [320]
                                const float* __restrict__ W_ss,   // [320,64,2]
                                const float* __restrict__ b_ss,   // [320]
                                float* __restrict__ out)          // [B,HID,H,W]
{
    __shared__ v2f   s_x[HDIM * XS2];              // [i][k/2]      (K-contiguous)
    __shared__ v2f   s_h[(HDIM + 1) * XS2];        // row 0 zeros; row i+1 = h[:, i]
    __shared__ float s_c[2][(HDIM + 1) * GS];      // row 0 zeros; row i+1 = c[:, i]
    __shared__ float s_g[GATES * GS];              // gate pre-activations [row][i]

    float* const s_xf = (float*)s_x;
    float* const s_hf = (float*)s_h;

    const int b    = blockIdx.x;
    const int tid  = threadIdx.x;
    const int wave = tid >> 5;            // 0..19 -> M-tile
    const int lane = tid & 31;
    const int l16  = lane & 15;
    const int hi   = lane >> 4;           // half-wave -> K offset +2
    const int mrow = wave * 16 + l16;     // gate row held by this lane in A

    // ---- preload combined A (weights) into registers, WMMA fp32 A layout ----
    // A[m][k]: lane = (k&2)*8 + m ; vgpr element = k&1
    v2f a_w[NKC];
#pragma unroll
    for (int kc = 0; kc < NKC; ++kc) {
#pragma unroll
        for (int v = 0; v < 2; ++v) {
            const int k = kc * 4 + hi * 2 + v;
            float w;
            if (k < CIN)            w = W_is[mrow * CIN + k];
            else if (k < CIN + HID) w = W_ss[mrow * (HID * 2) + (k - CIN) * 2 + 1];
            else                    w = W_ss[mrow * (HID * 2) + (k - CIN - HID) * 2 + 0];
            a_w[kc][v] = w;
        }
    }

    // bias folded into accumulator init; D layout row M = v + hi*8
    float biasv[8];
#pragma unroll
    for (int v = 0; v < 8; ++v) {
        const int r = wave * 16 + v + hi * 8;
        biasv[v] = b_is[r] + b_ss[r];
    }

    // zero-init h (incl. pad row) and both c buffers (incl. pad rows)
    for (int idx = tid; idx < (HDIM + 1) * XSF; idx += NTHREADS) s_hf[idx] = 0.0f;
    for (int idx = tid; idx < (HDIM + 1) * GS; idx += NTHREADS) {
        s_c[0][idx] = 0.0f; s_c[1][idx] = 0.0f;
    }

    const float* featB = feat + (size_t)b * CIN * HDIM * WDIM;
    float*       outB  = out  + (size_t)b * HID * HDIM * WDIM;

    float nx[NLOAD];
    gather_col(featB, 0, tid, nx);        // prologue: column 0
    __syncthreads();

    for (int col = 0; col < 2 * WDIM - 1; ++col) {
        // ---- phase A: commit prefetched skewed column to LDS (transposed) ----
#pragma unroll
        for (int t = 0; t < NLOAD; ++t) {
            const int idx = tid + t * NTHREADS;
            if (idx < CIN * HDIM) {
                const int c = idx & 63, i = idx >> 6;
                s_xf[i * XSF + c] = nx[t];     // lanes: consecutive c -> contiguous
            }
        }
        __syncthreads();

        // prefetch next column; latency overlaps the WMMA chain + gate math
        gather_col(featB, col + 1, tid, nx);

        // ---- phase B: g = A @ [x; h; h_shift] + bias (fp32 WMMA, K=192) ----
        v8f acc[NT];
#pragma unroll
        for (int n = 0; n < NT; ++n)
#pragma unroll
            for (int v = 0; v < 8; ++v) acc[n][v] = biasv[v];

#pragma unroll
        for (int kc = 0; kc < NKC; ++kc) {
            const int region = kc >> 4;            // 0: x, 1: h, 2: h shifted
            const int kh = (kc * 2 + hi) & 31;     // v2f index inside region row
#pragma unroll
            for (int n = 0; n < NT; ++n) {
                const int ci = n * 16 + l16;       // output column (H index)
                v2f bv;
                if (region == 0)      bv = s_x[ci * XS2 + kh];         // x[:, i]
                else if (region == 1) bv = s_h[(ci + 1) * XS2 + kh];   // h[:, i]
                else                  bv = s_h[ci * XS2 + kh];         // h[:, i-1]
                acc[n] = __builtin_amdgcn_wmma_f32_16x16x4_f32(
                    false, a_w[kc], false, bv, (short)0, acc[n], false, false);
            }
        }

        // store D tiles: row M = v + hi*8, col N = l16 (contiguous per half-wave)
#pragma unroll
        for (int n = 0; n < NT; ++n)
#pragma unroll
            for (int v = 0; v < 8; ++v)
                s_g[(wave * 16 + v + hi * 8) * GS + n * 16 + l16] = acc[n][v];
        __syncthreads();

        // ---- phase C: LSTM gate update + direct unskewed output scatter ----
        const int pr = col & 1;                    // c read-buffer parity
#pragma unroll
        for (int t = 0; t < NLOAD; ++t) {
            const int idx = tid + t * NTHREADS;
            if (idx < HID * HDIM) {
                const int j = idx & 63, i = idx >> 6;   // j fast -> contiguous LDS
                const float o  = s_g[(0   + j) * GS + i];
                const float fl = s_g[(64  + j) * GS + i];
                const float fu = s_g[(128 + j) * GS + i];
                const float ig = s_g[(192 + j) * GS + i];
                const float cg = s_g[(256 + j) * GS + i];
                const float c_old = s_c[pr][(i + 1) * GS + j];
                const float c_sh  = s_c[pr][i * GS + j];          // row 0 = zeros
                const float cn = sigf(fl) * c_old + sigf(fu) * c_sh
                               + sigf(ig) * tanhf(cg);
                const float hn = sigf(o) * tanhf(cn);
                s_c[pr ^ 1][(i + 1) * GS + j] = cn;
                s_hf[(i + 1) * XSF + j] = hn;
                const int w = col - i;
                if (w >= 0 && w < WDIM) outB[(j * HDIM + i) * WDIM + w] = hn;
            }
        }
        __syncthreads();
    }
}

extern "C" void kernel_launch(void* const* d_in, const int* in_sizes, int n_in,
                              void* d_out, int out_size, void* d_ws, size_t ws_size,
                              hipStream_t stream) {
    (void)n_in; (void)out_size; (void)d_ws; (void)ws_size;
    const float* feat = (const float*)d_in[0];
    const float* Wis  = (const float*)d_in[1];
    const float* bis  = (const float*)d_in[2];
    const float* Wss  = (const float*)d_in[3];
    const float* bss  = (const float*)d_in[4];
    float* outp = (float*)d_out;
    const int Bn = in_sizes[0] / (CIN * HDIM * WDIM);   // = 16
    diag_pixel_lstm<<<dim3(Bn), dim3(NTHREADS), 0, stream>>>(feat, Wis, bis, Wss, bss, outp);
}